// ScaleIndependentProjectionLoss_39513699123326
// MI455X (gfx1250) — compile-verified
//
#include <hip/hip_runtime.h>
#include <hip/hip_bf16.h>

typedef float v8f __attribute__((ext_vector_type(8)));
typedef float v2f __attribute__((ext_vector_type(2)));

#define B_TOTAL 65536
#define V_VIEWS 4
#define J_KP    32
#define WPB     8                      // waves per block
#define BLOCKS  (B_TOTAL / WPB)

#if __has_builtin(__builtin_amdgcn_rcpf)
#define FAST_RCP(x) __builtin_amdgcn_rcpf(x)
#else
#define FAST_RCP(x) (1.0f / (x))
#endif
#if __has_builtin(__builtin_amdgcn_rsqf)
#define FAST_RSQ(x) __builtin_amdgcn_rsqf(x)
#else
#define FAST_RSQ(x) (1.0f / sqrtf(x))
#endif

// ---------- cross-lane reduction primitives ----------
// ds_swizzle_b32 BITMODE: dst[lane] = src[((lane & and) | or) ^ xor]
template <int XM>
__device__ __forceinline__ float swz_xor(float x) {
  return __int_as_float(
      __builtin_amdgcn_ds_swizzle(__float_as_int(x), 0x1f | (XM << 10)));
}

#if __has_builtin(__builtin_amdgcn_update_dpp)
// DPP16 ROW_ROR:n (ctrl 0x120+n): rotate within each 16-lane row.
// Folds into a single v_add_f32_dpp (verified in round-4 codegen).
template <int N>
__device__ __forceinline__ float row_ror_f(float x) {
  return __int_as_float(__builtin_amdgcn_update_dpp(
      0, __float_as_int(x), 0x120 + N, 0xF, 0xF, true));
}
__device__ __forceinline__ float rot1(float x) { return row_ror_f<1>(x); }
__device__ __forceinline__ float rot2(float x) { return row_ror_f<2>(x); }
__device__ __forceinline__ float rot4(float x) { return row_ror_f<4>(x); }
__device__ __forceinline__ float rot8(float x) { return row_ror_f<8>(x); }
#else
// xor pairing is also a valid rotate-free fallback for sum reduction
__device__ __forceinline__ float rot1(float x) { return swz_xor<1>(x); }
__device__ __forceinline__ float rot2(float x) { return swz_xor<2>(x); }
__device__ __forceinline__ float rot4(float x) { return swz_xor<4>(x); }
__device__ __forceinline__ float rot8(float x) { return swz_xor<8>(x); }
#endif
// crossing 16-lane rows in wave32: DS swizzle xor-16
__device__ __forceinline__ float xlane16(float x) { return swz_xor<16>(x); }

// Rotation prefix-doubling: after ror 1,2,4,8 every lane holds its full
// 16-lane row sum; xor-16 then completes the wave32 sum (all lanes).
__device__ __forceinline__ float wave_reduce_add(float x) {
  x += rot1(x);
  x += rot2(x);
  x += rot4(x);
  x += rot8(x);
  x += xlane16(x);
  return x;
}

// Fused 2-value full reduction (interleaved chains hide DPP dep latency).
__device__ __forceinline__ void wave_reduce_add2(float& a, float& b) {
  a += rot1(a);     b += rot1(b);
  a += rot2(a);     b += rot2(b);
  a += rot4(a);     b += rot4(b);
  a += rot8(a);     b += rot8(b);
  a += xlane16(a);  b += xlane16(b);
}

// Row-sum only (each 16-lane half reduced independently).
__device__ __forceinline__ void half_reduce_add2(float& a, float& b) {
  a += rot1(a);     b += rot1(b);
  a += rot2(a);     b += rot2(b);
  a += rot4(a);     b += rot4(b);
  a += rot8(a);     b += rot8(b);
}

__device__ __forceinline__ float readlane_f(float x, int lane) {
  return __int_as_float(__builtin_amdgcn_readlane(__float_as_int(x), lane));
}

// One wave32 per batch element b. For each view v:
//   Xc = [X|1] (16x4) x [R^T;t] (4x16) via V_WMMA_F32_16X16X4_F32 (x2 for J=32).
//   z_bar comes analytically from R[2,:]. X_mean + t2 (X_mean reduced once per b),
//   so only one fused butterfly (|uv|^2, |ik|^2) remains per view; the L1 diff
//   is accumulated per-lane and reduced once after the view loop.
__global__ void __launch_bounds__(256)
proj_loss_kernel(const float* __restrict__ Kmat,   // (B,V,3,3)
                 const float* __restrict__ cam,    // (B,V,3,4) = [R|t]
                 const float* __restrict__ Xw,     // (B,J,3)
                 const float* __restrict__ ik,     // (B,V,J,2)
                 float* __restrict__ partial) {
  __shared__ float xcbuf[WPB * 64];    // per-wave: 2 components x 32 joints
  __shared__ float redbuf[WPB];

  const int lane = threadIdx.x & 31;
  const int wave = threadIdx.x >> 5;
  const int b    = blockIdx.x * WPB + wave;

  const int  m    = lane & 15;
  const bool hi   = lane >= 16;
  const int  jb   = hi ? 8 : 0;        // D-row base for this lane half
  const int  slot = wave * 64;

  // ---- A matrices (WMMA A-layout, 16x4 f32): built once per b, reused 4 views.
  // lanes 0-15: vgpr0=K0, vgpr1=K1 ; lanes 16-31: vgpr0=K2, vgpr1=K3(==1.0)
  const float* Xb = Xw + (size_t)b * (J_KP * 3);
  v2f a0, a1;
  {
    const int k0 = hi ? 2 : 0;
    a0.x = Xb[m * 3 + k0];
    a0.y = hi ? 1.0f : Xb[m * 3 + 1];
    a1.x = Xb[(m + 16) * 3 + k0];
    a1.y = hi ? 1.0f : Xb[(m + 16) * 3 + 1];
  }

  // ---- one-time joint sums: lo half accumulates (SumX0, SumX1), hi half
  // accumulates (SumX2, unused) straight from the A registers.
  float ax = a0.x + a1.x;
  float ay = a0.y + a1.y;
  half_reduce_add2(ax, ay);
  const float sx0 = readlane_f(ax, 0);    // sum_j X[j,0]
  const float sx1 = readlane_f(ay, 0);    // sum_j X[j,1]
  const float sx2 = readlane_f(ax, 16);   // sum_j X[j,2]

  float accd = 0.0f;                      // per-lane sum of |diff| over views
#pragma unroll
  for (int v = 0; v < V_VIEWS; ++v) {
    const size_t bv = (size_t)b * V_VIEWS + v;
    const float* camv = cam + bv * 12;

    // ---- B matrix (4x16 f32): B[k,i] = R[i,k], B[3,i] = t[i].
    // Row-striped layout: vgpr0 = rows {0,2}, vgpr1 = rows {1,3}.
    // cam row i = [R[i,0],R[i,1],R[i,2],t[i]] -> one aligned float2 load.
    const int    mc = (m < 3) ? m : 2;
    const float2 rv = *(const float2*)(camv + mc * 4 + (hi ? 2 : 0));
    v2f bm;
    bm.x = (m < 3) ? rv.x : 0.0f;
    bm.y = (m < 3) ? rv.y : 0.0f;

    // ---- initial keypoints early (coalesced float2, overlaps WMMA latency)
    const float2 ikv = *(const float2*)(ik + (bv * J_KP + lane) * 2);
    const float  i0 = ikv.x, i1 = ikv.y;

    v8f cz = {};
    v8f d0 = __builtin_amdgcn_wmma_f32_16x16x4_f32(
        false, a0, false, bm, (short)0, cz, false, false);   // joints 0..15
    v8f d1 = __builtin_amdgcn_wmma_f32_16x16x4_f32(
        false, a1, false, bm, (short)0, cz, false, false);   // joints 16..31

    // ---- analytic mean depth: z_bar = R[2,:].mean(X) + t2
    const float4 c2 = *(const float4*)(camv + 8);  // [R20,R21,R22,t2]
    const float  zbar =
        (c2.x * sx0 + c2.y * sx1 + c2.z * sx2) * (1.0f / 32.0f) + c2.w;
    const float invz = FAST_RCP(zbar);

    // ---- transpose D columns 0..1 (Xc x,y) into joint-per-lane via LDS.
    // D layout: vgpr r, lane L -> D[M = (L<16 ? r : r+8)][N = L&15]
    if (m < 2) {
      float* dst = xcbuf + slot + m * J_KP + jb;
      *(float4*)(dst +  0) = make_float4(d0[0], d0[1], d0[2], d0[3]);
      *(float4*)(dst +  4) = make_float4(d0[4], d0[5], d0[6], d0[7]);
      *(float4*)(dst + 16) = make_float4(d1[0], d1[1], d1[2], d1[3]);
      *(float4*)(dst + 20) = make_float4(d1[4], d1[5], d1[6], d1[7]);
    }
    // LDS is in-order per wave; wait for the DS stores, block compiler reordering.
    asm volatile("s_wait_dscnt 0x0" ::: "memory");

    const float xy0 = xcbuf[slot + lane] * invz;        // Xc.x / z_bar
    const float xy1 = xcbuf[slot + 32 + lane] * invz;   // Xc.y / z_bar

    // ---- intrinsics (full 2x2 + principal point)
    const float* Kp = Kmat + bv * 9;
    const float uv0 = Kp[0] * xy0 + Kp[1] * xy1 + Kp[2];
    const float uv1 = Kp[3] * xy0 + Kp[4] * xy1 + Kp[5];

    // ---- the only per-view reduction: fused |uv|^2 and |ik|^2
    float sp  = uv0 * uv0 + uv1 * uv1;
    float siv = i0 * i0 + i1 * i1;
    wave_reduce_add2(sp, siv);

    const float rp = FAST_RSQ(sp);
    const float ri = FAST_RSQ(siv);
    accd += fabsf(uv0 * rp - i0 * ri) + fabsf(uv1 * rp - i1 * ri);
  }

  // ---- single deferred reduction of the L1 terms (sum_v sum_{j,c})
  const float dsum = wave_reduce_add(accd);
  if (lane == 0) redbuf[wave] = dsum * (1.0f / 64.0f);
  __syncthreads();
  if (threadIdx.x == 0) {
    float s = 0.0f;
#pragma unroll
    for (int w = 0; w < WPB; ++w) s += redbuf[w];
    partial[blockIdx.x] = s;
  }
}

// Deterministic final reduction of block partials -> scalar mean.
__global__ void __launch_bounds__(256)
reduce_kernel(const float* __restrict__ partial, float* __restrict__ out, int n) {
  __shared__ float sbuf[256];
  float s = 0.0f;
  for (int i = threadIdx.x; i < n; i += 256) s += partial[i];
  sbuf[threadIdx.x] = s;
  __syncthreads();
  for (int stride = 128; stride > 0; stride >>= 1) {
    if ((int)threadIdx.x < stride) sbuf[threadIdx.x] += sbuf[threadIdx.x + stride];
    __syncthreads();
  }
  if (threadIdx.x == 0)
    out[0] = sbuf[0] * (1.0f / (float)((size_t)B_TOTAL * V_VIEWS));
}

extern "C" void kernel_launch(void* const* d_in, const int* in_sizes, int n_in,
                              void* d_out, int out_size, void* d_ws, size_t ws_size,
                              hipStream_t stream) {
  const float* Kmat = (const float*)d_in[0];  // (B,V,3,3)
  const float* cam  = (const float*)d_in[1];  // (B,V,3,4)
  const float* Xw   = (const float*)d_in[2];  // (B,J,3)
  const float* ik   = (const float*)d_in[3];  // (B,V,J,2)
  float* out = (float*)d_out;
  float* partial = (float*)d_ws;              // BLOCKS floats (32 KB)

  proj_loss_kernel<<<dim3(BLOCKS), dim3(256), 0, stream>>>(Kmat, cam, Xw, ik, partial);
  reduce_kernel<<<dim3(1), dim3(256), 0, stream>>>(partial, out, BLOCKS);
}